// SparseFCN_73426760893125
// MI455X (gfx1250) — compile-verified
//
#include <hip/hip_runtime.h>

// ---------------------------------------------------------------------------
// MI455X (gfx1250) 4-layer strided/dilated conv pipeline, fp32 WMMA.
//
// Memory-bound (AI ~9 FLOP/B << machine balance), so V_WMMA_F32_16X16X4_F32
// gives reference-exact fp32 MACs at zero perf cost. One wave32 per
// 16-pixel x 16-channel output tile. Weight (B) fragments and biases are
// zero-padded and staged in LDS in fragment order -> aligned ds_loads, no
// lane divergence anywhere in the per-wave path. A-tiles are loaded
// branch-free (clamped address + v_cndmask zeroing) and batched per kernel
// row so a clause of global loads pipelines ahead of the WMMA burst.
// Transposed convs exploit stride=dilation=2: odd output pixels are pure
// bias (vector pre-fill), only the even-even quarter is gathered.
// ---------------------------------------------------------------------------

typedef __attribute__((ext_vector_type(2))) float v2f;
typedef __attribute__((ext_vector_type(4))) float v4f;
typedef __attribute__((ext_vector_type(8))) float v8f;

__device__ __forceinline__ v8f wmma4(v2f a, v2f b, v8f c) {
#if defined(__gfx1250__)
  return __builtin_amdgcn_wmma_f32_16x16x4_f32(false, a, false, b, (short)0, c,
                                               false, false);
#else
  (void)a; (void)b;
  return c;  // host-pass stub
#endif
}

__device__ __forceinline__ float relu_(float v) { return v > 0.f ? v : 0.f; }

__device__ __forceinline__ v2f sel2(v2f a, bool p) {
  v2f r; r.x = p ? a.x : 0.f; r.y = p ? a.y : 0.f; return r;
}
__device__ __forceinline__ v4f sel4(v4f a, bool p) {
  v4f r;
  r.x = p ? a.x : 0.f; r.y = p ? a.y : 0.f;
  r.z = p ? a.z : 0.f; r.w = p ? a.w : 0.f;
  return r;
}

// ---------------------------------------------------------------------------
// Layer 1: x (8,1024,1024,4) -> x1 (8,510,510,8). K/tap = 4 (1 WMMA/tap).
// ---------------------------------------------------------------------------
__global__ __launch_bounds__(256) void conv1_wmma(
    const float* __restrict__ x, const float* __restrict__ w1,
    const float* __restrict__ b1, float* __restrict__ x1) {
  constexpr int B = 8, H = 1024, W = 1024, CI = 4;
  constexpr int OH = 510, OW = 510, CO = 8;
  constexpr int TPR = 32;  // ceil(510/16)
  __shared__ float lb[9 * 64];   // B fragments, N zero-padded to 16
  __shared__ float lbias[16];    // bias, zero-padded to 16
  for (int i = threadIdx.x; i < 9 * 64; i += 256) {
    const int tap = i >> 6, r = i & 63, ln = r >> 1, v = r & 1;
    const int gg = ln >> 4, nn = ln & 15;
    lb[i] = (nn < CO) ? w1[(tap * CI + 2 * gg + v) * CO + nn] : 0.f;
  }
  if (threadIdx.x < 16)
    lbias[threadIdx.x] = (threadIdx.x < CO) ? b1[threadIdx.x] : 0.f;
  __syncthreads();

  const int wave = (int)((blockIdx.x * blockDim.x + threadIdx.x) >> 5);
  const int lane = threadIdx.x & 31;
  if (wave >= B * OH * TPR) return;
  const int ow0 = (wave % TPR) * 16;
  const int oh  = (wave / TPR) % OH;
  const int b   = wave / (TPR * OH);
  const int g = lane >> 4, n = lane & 15;
  const bool valid = (ow0 + n) < OW;
  const int owc = valid ? (ow0 + n) : (OW - 1);  // clamped -> safe load

  v8f acc = {};
#pragma unroll
  for (int kh = 0; kh < 3; ++kh) {
    const float* row = x + (long)(b * H + 2 * oh + 2 * kh) * (W * CI);
    v2f a[3];
#pragma unroll
    for (int kw = 0; kw < 3; ++kw)
      a[kw] = *(const v2f*)&row[(2 * owc + 2 * kw) * CI + 2 * g];
#pragma unroll
    for (int kw = 0; kw < 3; ++kw) {
      const v2f bw = *(const v2f*)&lb[(kh * 3 + kw) * 64 + lane * 2];
      acc = wmma4(sel2(a[kw], valid), bw, acc);
    }
  }
  const float bias = lbias[n];
  float* orow = x1 + (long)(b * OH + oh) * OW * CO;
  if (ow0 + 16 <= OW) {
    if (n < CO) {
#pragma unroll
      for (int v = 0; v < 8; ++v)
        orow[(ow0 + v + 8 * g) * CO + n] = relu_(acc[v] + bias);
    }
  } else {
#pragma unroll
    for (int v = 0; v < 8; ++v) {
      const int ow = ow0 + v + 8 * g;
      if (n < CO && ow < OW) orow[ow * CO + n] = relu_(acc[v] + bias);
    }
  }
}

// ---------------------------------------------------------------------------
// Layer 2: x1 (8,510,510,8) -> x2 (8,253,253,16). K/tap = 8 (2 WMMAs/tap).
// K-slot mapping co = g*4 + 2s + v -> one b128 A load per tap per lane.
// ---------------------------------------------------------------------------
__global__ __launch_bounds__(256) void conv2_wmma(
    const float* __restrict__ x1, const float* __restrict__ w2,
    const float* __restrict__ b2, float* __restrict__ x2) {
  constexpr int B = 8, H = 510, W = 510, CI = 8;
  constexpr int OH = 253, OW = 253, CO = 16;
  constexpr int TPR = 16;  // ceil(253/16)
  __shared__ float lb[9 * 128];
  __shared__ float lbias[16];
  for (int i = threadIdx.x; i < 9 * 128; i += 256) {
    const int tap = i >> 7, r = i & 127, ln = r >> 2, j = r & 3;
    const int gg = ln >> 4, nn = ln & 15;
    lb[i] = w2[(tap * CI + gg * 4 + j) * CO + nn];
  }
  if (threadIdx.x < 16) lbias[threadIdx.x] = b2[threadIdx.x];
  __syncthreads();

  const int wave = (int)((blockIdx.x * blockDim.x + threadIdx.x) >> 5);
  const int lane = threadIdx.x & 31;
  if (wave >= B * OH * TPR) return;
  const int ow0 = (wave % TPR) * 16;
  const int oh  = (wave / TPR) % OH;
  const int b   = wave / (TPR * OH);
  const int g = lane >> 4, n = lane & 15;
  const bool valid = (ow0 + n) < OW;
  const int owc = valid ? (ow0 + n) : (OW - 1);

  v8f acc = {};
#pragma unroll
  for (int kh = 0; kh < 3; ++kh) {
    const float* row = x1 + (long)(b * H + 2 * oh + 2 * kh) * (W * CI);
    v4f a[3];
#pragma unroll
    for (int kw = 0; kw < 3; ++kw)
      a[kw] = *(const v4f*)&row[(2 * owc + 2 * kw) * CI + g * 4];
#pragma unroll
    for (int kw = 0; kw < 3; ++kw) {
      const v4f aa = sel4(a[kw], valid);
      const v4f bw = *(const v4f*)&lb[(kh * 3 + kw) * 128 + lane * 4];
      v2f a0 = {aa.x, aa.y}, a1 = {aa.z, aa.w};
      v2f b0 = {bw.x, bw.y}, b1v = {bw.z, bw.w};
      acc = wmma4(a0, b0, acc);
      acc = wmma4(a1, b1v, acc);
    }
  }
  const float bias = lbias[n];
  float* orow = x2 + (long)(b * OH + oh) * OW * CO;
  if (ow0 + 16 <= OW) {
#pragma unroll
    for (int v = 0; v < 8; ++v)
      orow[(ow0 + v + 8 * g) * CO + n] = relu_(acc[v] + bias);
  } else {
#pragma unroll
    for (int v = 0; v < 8; ++v) {
      const int ow = ow0 + v + 8 * g;
      if (ow < OW) orow[ow * CO + n] = relu_(acc[v] + bias);
    }
  }
}

// ---------------------------------------------------------------------------
// Vectorized bias pre-fill: odd pixels of the stride-2/dil-2 transposed convs
// get exactly (relu of) bias. 16B stores.
// ---------------------------------------------------------------------------
__global__ __launch_bounds__(256) void fill_bias4(
    float* __restrict__ out, const float* __restrict__ bias, int cmask,
    int relu_flag, long n4) {
  const long q = (long)blockIdx.x * blockDim.x + threadIdx.x;
  if (q >= n4) return;
  const long base = q * 4;
  v4f v;
  v.x = bias[(base + 0) & cmask];
  v.y = bias[(base + 1) & cmask];
  v.z = bias[(base + 2) & cmask];
  v.w = bias[(base + 3) & cmask];
  if (relu_flag) {
    v.x = relu_(v.x); v.y = relu_(v.y); v.z = relu_(v.z); v.w = relu_(v.w);
  }
  *(v4f*)&out[base] = v;
}

// ---------------------------------------------------------------------------
// Layer 3: x3 = relu(convT(x2, wi2)+bi2) on even-even pixels (p=2e, q=e-k).
// x2 (8,253,253,16) -> x3 (8,510,510,8). K/tap = 16 (4 WMMAs/tap).
// K-slot mapping co = g*8 + 2s + v -> two b128 A loads per tap per lane.
// ---------------------------------------------------------------------------
__global__ __launch_bounds__(256) void invconv2_wmma(
    const float* __restrict__ x2, const float* __restrict__ wi2,
    const float* __restrict__ bi2, float* __restrict__ x3) {
  constexpr int B = 8, QH = 253, QW = 253, CIN = 16;
  constexpr int PH = 510, PW = 510, CO = 8;
  constexpr int EW = 255;  // even coords e in [0,254]
  constexpr int TPR = 16;  // ceil(255/16)
  __shared__ float lb[9 * 256];
  __shared__ float lbias[16];
  for (int i = threadIdx.x; i < 9 * 256; i += 256) {
    const int tap = i >> 8, r = i & 255, ln = r >> 3, j = r & 7;
    const int gg = ln >> 4, nn = ln & 15;
    lb[i] = (nn < CO) ? wi2[(tap * CO + nn) * CIN + gg * 8 + j] : 0.f;
  }
  if (threadIdx.x < 16)
    lbias[threadIdx.x] = (threadIdx.x < CO) ? bi2[threadIdx.x] : 0.f;
  __syncthreads();

  const int wave = (int)((blockIdx.x * blockDim.x + threadIdx.x) >> 5);
  const int lane = threadIdx.x & 31;
  if (wave >= B * EW * TPR) return;
  const int pw0 = (wave % TPR) * 16;
  const int eh  = (wave / TPR) % EW;
  const int b   = wave / (TPR * EW);
  const int g = lane >> 4, n = lane & 15;
  const bool evalid = (pw0 + n) < EW;

  v8f acc = {};
#pragma unroll
  for (int kh = 0; kh < 3; ++kh) {
    const int qh = eh - kh;
    if ((unsigned)qh >= (unsigned)QH) continue;  // wave-uniform tap-row skip
    const float* row = x2 + (long)(b * QH + qh) * (QW * CIN);
    v4f a0[3], a1[3];
    bool av[3];
#pragma unroll
    for (int kw = 0; kw < 3; ++kw) {
      const int qw = pw0 + n - kw;
      av[kw] = evalid && ((unsigned)qw < (unsigned)QW);
      int qwc = qw < 0 ? 0 : (qw > QW - 1 ? QW - 1 : qw);
      a0[kw] = *(const v4f*)&row[qwc * CIN + g * 8];
      a1[kw] = *(const v4f*)&row[qwc * CIN + g * 8 + 4];
    }
#pragma unroll
    for (int kw = 0; kw < 3; ++kw) {
      const v4f xa = sel4(a0[kw], av[kw]);
      const v4f xb = sel4(a1[kw], av[kw]);
      const float* bp = &lb[(kh * 3 + kw) * 256 + lane * 8];
      const v4f bw0 = *(const v4f*)bp;
      const v4f bw1 = *(const v4f*)(bp + 4);
      v2f p0 = {xa.x, xa.y}, p1 = {xa.z, xa.w};
      v2f p2 = {xb.x, xb.y}, p3 = {xb.z, xb.w};
      v2f q0 = {bw0.x, bw0.y}, q1 = {bw0.z, bw0.w};
      v2f q2 = {bw1.x, bw1.y}, q3 = {bw1.z, bw1.w};
      acc = wmma4(p0, q0, acc);
      acc = wmma4(p1, q1, acc);
      acc = wmma4(p2, q2, acc);
      acc = wmma4(p3, q3, acc);
    }
  }
  const float bias = lbias[n];
  float* orow = x3 + (long)(b * PH + 2 * eh) * PW * CO;
  if (pw0 + 16 <= EW) {
    if (n < CO) {
#pragma unroll
      for (int v = 0; v < 8; ++v)
        orow[2 * (pw0 + v + 8 * g) * CO + n] = relu_(acc[v] + bias);
    }
  } else {
#pragma unroll
    for (int v = 0; v < 8; ++v) {
      const int e = pw0 + v + 8 * g;
      if (n < CO && e < EW) orow[2 * e * CO + n] = relu_(acc[v] + bias);
    }
  }
}

// ---------------------------------------------------------------------------
// Layer 4: y = convT(x3, wi1)+bi1 on even-even pixels.
// x3 (8,510,510,8) -> y (8,1024,1024,2). K/tap = 8 (2 WMMAs/tap).
// ---------------------------------------------------------------------------
__global__ __launch_bounds__(256) void invconv1_wmma(
    const float* __restrict__ x3, const float* __restrict__ wi1,
    const float* __restrict__ bi1, float* __restrict__ y) {
  constexpr int B = 8, QH = 510, QW = 510, CIN = 8;
  constexpr int PH = 1024, PW = 1024, CO = 2;
  constexpr int EW = 512;  // even coords e in [0,511]
  constexpr int TPR = 32;  // 512/16, all tiles full
  __shared__ float lb[9 * 128];
  __shared__ float lbias[16];
  for (int i = threadIdx.x; i < 9 * 128; i += 256) {
    const int tap = i >> 7, r = i & 127, ln = r >> 2, j = r & 3;
    const int gg = ln >> 4, nn = ln & 15;
    lb[i] = (nn < CO) ? wi1[(tap * CO + nn) * CIN + gg * 4 + j] : 0.f;
  }
  if (threadIdx.x < 16)
    lbias[threadIdx.x] = (threadIdx.x < CO) ? bi1[threadIdx.x] : 0.f;
  __syncthreads();

  const int wave = (int)((blockIdx.x * blockDim.x + threadIdx.x) >> 5);
  const int lane = threadIdx.x & 31;
  if (wave >= B * EW * TPR) return;
  const int pw0 = (wave % TPR) * 16;
  const int eh  = (wave / TPR) % EW;
  const int b   = wave / (TPR * EW);
  const int g = lane >> 4, n = lane & 15;

  v8f acc = {};
#pragma unroll
  for (int kh = 0; kh < 3; ++kh) {
    const int qh = eh - kh;
    if ((unsigned)qh >= (unsigned)QH) continue;
    const float* row = x3 + (long)(b * QH + qh) * (QW * CIN);
    v4f a[3];
    bool av[3];
#pragma unroll
    for (int kw = 0; kw < 3; ++kw) {
      const int qw = pw0 + n - kw;  // pw0+n < 512 always
      av[kw] = (unsigned)qw < (unsigned)QW;
      int qwc = qw < 0 ? 0 : (qw > QW - 1 ? QW - 1 : qw);
      a[kw] = *(const v4f*)&row[qwc * CIN + g * 4];
    }
#pragma unroll
    for (int kw = 0; kw < 3; ++kw) {
      const v4f aa = sel4(a[kw], av[kw]);
      const v4f bw = *(const v4f*)&lb[(kh * 3 + kw) * 128 + lane * 4];
      v2f a0 = {aa.x, aa.y}, a1 = {aa.z, aa.w};
      v2f b0 = {bw.x, bw.y}, b1v = {bw.z, bw.w};
      acc = wmma4(a0, b0, acc);
      acc = wmma4(a1, b1v, acc);
    }
  }
  const float bias = lbias[n];
  float* orow = y + (long)(b * PH + 2 * eh) * PW * CO;
  if (n < CO) {
#pragma unroll
    for (int v = 0; v < 8; ++v)
      orow[2 * (pw0 + v + 8 * g) * CO + n] = acc[v] + bias;
  }
}

// ---------------------------------------------------------------------------
extern "C" void kernel_launch(void* const* d_in, const int* in_sizes, int n_in,
                              void* d_out, int out_size, void* d_ws,
                              size_t ws_size, hipStream_t stream) {
  (void)in_sizes; (void)n_in; (void)out_size; (void)ws_size;
  const float* x   = (const float*)d_in[0];
  const float* w1  = (const float*)d_in[1];
  const float* b1  = (const float*)d_in[2];
  const float* w2  = (const float*)d_in[3];
  const float* b2  = (const float*)d_in[4];
  const float* wi2 = (const float*)d_in[5];
  const float* bi2 = (const float*)d_in[6];
  const float* wi1 = (const float*)d_in[7];
  const float* bi1 = (const float*)d_in[8];
  float* y = (float*)d_out;

  // Workspace layout (fp32): x1 | x2 | x3  (~158 MiB total)
  const long n_x1 = 8L * 510 * 510 * 8;    // 16,646,400
  const long n_x2 = 8L * 253 * 253 * 16;   //  8,193,152
  const long n_x3 = n_x1;
  const long n_y  = 8L * 1024 * 1024 * 2;  // 16,777,216
  float* x1 = (float*)d_ws;
  float* x2 = x1 + n_x1;
  float* x3 = x2 + n_x2;

  const dim3 blk(256);  // 8 wave32s per block
  const int WPB = 256 / 32;

  {  // Layer 1
    const long tiles = 8L * 510 * 32;
    conv1_wmma<<<dim3((unsigned)((tiles + WPB - 1) / WPB)), blk, 0, stream>>>(
        x, w1, b1, x1);
  }
  {  // Layer 2
    const long tiles = 8L * 253 * 16;
    conv2_wmma<<<dim3((unsigned)((tiles + WPB - 1) / WPB)), blk, 0, stream>>>(
        x1, w2, b2, x2);
  }
  {  // Layer 3: bias fill, then gather-GEMM over even-even pixels
    fill_bias4<<<dim3((unsigned)(n_x3 / 4 + 255) / 256), blk, 0, stream>>>(
        x3, bi2, 7, 1, n_x3 / 4);
    const long tiles = 8L * 255 * 16;
    invconv2_wmma<<<dim3((unsigned)((tiles + WPB - 1) / WPB)), blk, 0,
                    stream>>>(x2, wi2, bi2, x3);
  }
  {  // Layer 4
    fill_bias4<<<dim3((unsigned)(n_y / 4 + 255) / 256), blk, 0, stream>>>(
        y, bi1, 1, 0, n_y / 4);
    const long tiles = 8L * 512 * 32;
    invconv1_wmma<<<dim3((unsigned)((tiles + WPB - 1) / WPB)), blk, 0,
                    stream>>>(x3, wi1, bi1, y);
  }
}